// SpatioAttention_24026047053979
// MI455X (gfx1250) — compile-verified
//
#include <hip/hip_runtime.h>
#include <hip/hip_bf16.h>

// ---------------------------------------------------------------------------
// SpatioAttention for MI455X (gfx1250), wave32.
//
// QKV projections (70% of FLOPs) run on V_WMMA_F32_16X16X4_F32 (A 16x4 f32,
// B 4x16 f32, C/D 16x16 f32): GEMM rows = (token,neighbor) pairs, inner dim
// = DIM_HEAD = 4, columns = 32 head-channels; bias rides in as C.
// Input block staged to LDS with one coalesced read; output block staged to
// LDS and written with one coalesced store (18-float rows are otherwise
// badly strided). Softmax / AV / output proj are cheap VALU work.
// ---------------------------------------------------------------------------

typedef __attribute__((ext_vector_type(2))) float v2f;
typedef __attribute__((ext_vector_type(8))) float v8f;

#define T_TOK   16              // tokens per workgroup
#define NNB     10              // neighbors
#define ROWS    (T_TOK * NNB)   // 160 projection rows per workgroup
#define KV_STR  36              // LDS row stride (floats): 144B, 16B-aligned
#define ENC     54
#define SPLIT   14
#define OUTC    18

__device__ __forceinline__ v8f wmma_f32_16x16x4(v2f a, v2f b, v8f c) {
    // (neg_a, A, neg_b, B, c_mod, C, reuse_a, reuse_b)
    return __builtin_amdgcn_wmma_f32_16x16x4_f32(0, a, 0, b, (short)0, c, 0, 0);
}

__global__ __launch_bounds__(128)
void spatio_attn_kernel(const float* __restrict__ x_enc,
                        const float* __restrict__ Wq, const float* __restrict__ bq,
                        const float* __restrict__ Wk, const float* __restrict__ bk,
                        const float* __restrict__ Wv, const float* __restrict__ bv,
                        const float* __restrict__ Wo, const float* __restrict__ bo,
                        float* __restrict__ out, int nTok)
{
    __shared__ float Xls[T_TOK * ENC];      // staged input block   (3456 B)
    __shared__ float Kls[ROWS * KV_STR];    //                      (23040 B)
    __shared__ float Vls[ROWS * KV_STR];    //                      (23040 B)
    __shared__ float Qls[T_TOK * 32];       //                      (2048 B)
    __shared__ float Ols[T_TOK * 32];       //                      (2048 B)
    __shared__ float Obuf[T_TOK * OUTC];    // staged output block  (1152 B)

    const int tid  = threadIdx.x;
    const int wave = tid >> 5;
    const int lane = tid & 31;
    const int half = lane >> 4;      // 0 -> K={0,1}, 1 -> K={2,3}
    const int nloc = lane & 15;      // B column within block / A row
    const int tok0 = blockIdx.x * T_TOK;

    // ---------------- Phase 0: coalesced input staging -----------------------
    {
        const size_t base  = (size_t)tok0 * ENC;
        const size_t total = (size_t)nTok * ENC;
#pragma unroll
        for (int idx = tid; idx < T_TOK * ENC; idx += 128) {
            const size_t g = base + idx;
            Xls[idx] = (g < total) ? x_enc[g] : 0.0f;
        }
    }

    // ---------------- Phase 1 setup: B operands + bias-splat C ---------------
    // B[k][n] = W[n*4 + k]  (W flat over (head,e,d): col n = h*4+e, k = d)
    // lane holds (K = 2*half, N = nloc + 16*blk) and K+1  -> one float2 load.
    v2f Bq[2], Bk[2], Bv[2];
    v8f Cq[2], Ck[2], Cv[2];
#pragma unroll
    for (int blk = 0; blk < 2; ++blk) {
        const int n   = blk * 16 + nloc;
        const int off = n * 4 + 2 * half;
        Bq[blk][0] = Wq[off]; Bq[blk][1] = Wq[off + 1];
        Bk[blk][0] = Wk[off]; Bk[blk][1] = Wk[off + 1];
        Bv[blk][0] = Wv[off]; Bv[blk][1] = Wv[off + 1];
        const float vq = bq[n], vk = bk[n], vv = bv[n];
        Cq[blk] = (v8f){vq, vq, vq, vq, vq, vq, vq, vq};
        Ck[blk] = (v8f){vk, vk, vk, vk, vk, vk, vk, vk};
        Cv[blk] = (v8f){vv, vv, vv, vv, vv, vv, vv, vv};
    }

    __syncthreads();

    // ---------------- Phase 1: K/V projections via WMMA ----------------------
    // 10 row-tiles of 16 (token,neighbor) rows; wave-uniform loop bounds.
    for (int tile = wave; tile < ROWS / 16; tile += 4) {
        const int row = tile * 16 + nloc;        // A-matrix row M = lane&15
        const int tl  = row / NNB;               // local token
        const int j   = row - tl * NNB;          // neighbor
        const float* pa = &Xls[tl * ENC + SPLIT + j * 4 + 2 * half];
        v2f a; a[0] = pa[0]; a[1] = pa[1];

        v8f dk0 = wmma_f32_16x16x4(a, Bk[0], Ck[0]);
        v8f dk1 = wmma_f32_16x16x4(a, Bk[1], Ck[1]);
        v8f dv0 = wmma_f32_16x16x4(a, Bv[0], Cv[0]);
        v8f dv1 = wmma_f32_16x16x4(a, Bv[1], Cv[1]);

        // D layout: VGPR r, lane l -> (M = r + 8*(l>=16), N = l&15)
#pragma unroll
        for (int r = 0; r < 8; ++r) {
            const int lrow = tile * 16 + r + half * 8;
            Kls[lrow * KV_STR +      nloc] = dk0[r];
            Kls[lrow * KV_STR + 16 + nloc] = dk1[r];
            Vls[lrow * KV_STR +      nloc] = dv0[r];
            Vls[lrow * KV_STR + 16 + nloc] = dv1[r];
        }
    }

    // Q tile: 16 tokens, neighbor 0 only. Wave 3 (least loaded; uniform branch).
    if (wave == 3) {
        const float* pa = &Xls[nloc * ENC + SPLIT + 2 * half];
        v2f a; a[0] = pa[0]; a[1] = pa[1];
        v8f dq0 = wmma_f32_16x16x4(a, Bq[0], Cq[0]);
        v8f dq1 = wmma_f32_16x16x4(a, Bq[1], Cq[1]);
#pragma unroll
        for (int r = 0; r < 8; ++r) {
            const int t = r + half * 8;
            Qls[t * 32 +      nloc] = dq0[r];
            Qls[t * 32 + 16 + nloc] = dq1[r];
        }
    }

    __syncthreads();

    // ---------------- Phase 2: per-(token, head) attention -------------------
    {
        const int t = tid >> 3;          // 0..15
        const int h = tid & 7;           // 0..7
        const float4 q = *(const float4*)&Qls[t * 32 + h * 4];

        float s[NNB];
        float mx = -3.0e38f;
#pragma unroll
        for (int j = 0; j < NNB; ++j) {
            const float4 k = *(const float4*)&Kls[(t * NNB + j) * KV_STR + h * 4];
            s[j] = q.x * k.x + q.y * k.y + q.z * k.z + q.w * k.w;
            mx = fmaxf(mx, s[j]);
        }
        float sum = 0.0f;
#pragma unroll
        for (int j = 0; j < NNB; ++j) { s[j] = __expf(s[j] - mx); sum += s[j]; }
        const float inv = 1.0f / sum;

        float ox = 0.f, oy = 0.f, oz = 0.f, ow = 0.f;
#pragma unroll
        for (int j = 0; j < NNB; ++j) {
            const float4 v = *(const float4*)&Vls[(t * NNB + j) * KV_STR + h * 4];
            ox += s[j] * v.x; oy += s[j] * v.y; oz += s[j] * v.z; ow += s[j] * v.w;
        }
        float4 o; o.x = ox * inv; o.y = oy * inv; o.z = oz * inv; o.w = ow * inv;
        *(float4*)&Ols[t * 32 + h * 4] = o;
    }

    __syncthreads();

    // ---------------- Phase 3: output projection + staging -------------------
    if (tid < T_TOK * 4) {
        const int t = tid >> 2, c = tid & 3;
        float acc = bo[c];
#pragma unroll
        for (int m = 0; m < 32; ++m) acc += Ols[t * 32 + m] * Wo[c * 32 + m];
        Obuf[t * OUTC + SPLIT + c] = acc;
    }
    {   // passthrough channels 0..13 (no divisions: 8 lanes x {c, c+8})
        const int t = tid >> 3, c = tid & 7;
        Obuf[t * OUTC + c] = Xls[t * ENC + c];
        if (c + 8 < SPLIT) Obuf[t * OUTC + c + 8] = Xls[t * ENC + c + 8];
    }

    __syncthreads();

    // ---------------- Phase 4: coalesced output store ------------------------
    {
        const size_t base  = (size_t)tok0 * OUTC;
        const size_t total = (size_t)nTok * OUTC;
#pragma unroll
        for (int idx = tid; idx < T_TOK * OUTC; idx += 128) {
            const size_t g = base + idx;
            if (g < total) out[g] = Obuf[idx];
        }
    }
}

extern "C" void kernel_launch(void* const* d_in, const int* in_sizes, int n_in,
                              void* d_out, int out_size, void* d_ws, size_t ws_size,
                              hipStream_t stream) {
    const float* x_enc = (const float*)d_in[0];
    const float* Wq    = (const float*)d_in[1];
    const float* bq    = (const float*)d_in[2];
    const float* Wk    = (const float*)d_in[3];
    const float* bk    = (const float*)d_in[4];
    const float* Wv    = (const float*)d_in[5];
    const float* bv    = (const float*)d_in[6];
    const float* Wo    = (const float*)d_in[7];
    const float* bo    = (const float*)d_in[8];
    float* out = (float*)d_out;

    const int nTok = in_sizes[0] / ENC;                 // 256*2048 tokens
    const int grid = (nTok + T_TOK - 1) / T_TOK;        // 16 tokens per WG
    spatio_attn_kernel<<<grid, 128, 0, stream>>>(x_enc, Wq, bq, Wk, bk,
                                                 Wv, bv, Wo, bo, out, nTok);
}